// SELayer_50921132261525
// MI455X (gfx1250) — compile-verified
//
#include <hip/hip_runtime.h>

// ---------------------------------------------------------------------------
// SE layer over segment-pooled features, CDNA5 (gfx1250, wave32).
//  feats [N,256] f32, batch_idx [N] i32 (sorted, 32 segments),
//  W1 [256,16], b1[16], W2 [16,256], b2[256]  ->  out [N,256] f32
// Memory-bound: ~3.1 GB HBM traffic -> ~135us @ 23.3 TB/s.
// Tiny MLP done with V_WMMA_F32_16X16X4_F32 (full f32 precision).
// ---------------------------------------------------------------------------

#define NSEG   32
#define C      256
#define CR     16      // C / R
#define SPLITS 64      // deterministic split-reduction width per segment

typedef float v2f __attribute__((ext_vector_type(2)));
typedef float v8f __attribute__((ext_vector_type(8)));

// ---------------------------------------------------------------------------
// Kernel 1: per-(segment, split) partial sums. Deterministic (no FP atomics).
// grid = NSEG*SPLITS blocks, 256 threads (thread = channel).
// ---------------------------------------------------------------------------
__device__ __forceinline__ int lower_bound_i32(const int* __restrict__ a, int n, int v) {
    int lo = 0, hi = n;
    while (lo < hi) {
        int mid = (lo + hi) >> 1;
        if (a[mid] < v) lo = mid + 1; else hi = mid;
    }
    return lo;
}

__global__ void pool_partial_kernel(const float* __restrict__ feats,
                                    const int* __restrict__ bidx, int N,
                                    float* __restrict__ partial,   // [NSEG][SPLITS][C]
                                    int* __restrict__ counts) {    // [NSEG][SPLITS]
    const int seg   = blockIdx.x / SPLITS;
    const int split = blockIdx.x % SPLITS;
    // segment bounds via binary search in the sorted index array (uniform per block)
    const int start = lower_bound_i32(bidx, N, seg);
    const int end   = lower_bound_i32(bidx, N, seg + 1);
    const long len  = (long)(end - start);
    const int cb = start + (int)(len * split / SPLITS);
    const int ce = start + (int)(len * (split + 1) / SPLITS);

    const int c = threadIdx.x;               // channel
    float s = 0.0f;
    for (int r = cb; r < ce; ++r) {
        __builtin_prefetch(&feats[(size_t)(r + 16) * C + c], 0, 0);  // global_prefetch_b8
        s += feats[(size_t)r * C + c];
    }
    partial[((size_t)seg * SPLITS + split) * C + c] = s;
    if (c == 0) counts[seg * SPLITS + split] = ce - cb;
}

// ---------------------------------------------------------------------------
// Kernel 2: reduce partials -> pooled (LDS), then
//   h = relu(pooled @ W1 + b1)   via WMMA f32 16x16x4, waves 0..1 (M-tiles)
//   y = sigmoid(h @ W2 + b2)     via WMMA f32 16x16x4, 8 waves x 4 tiles
// 1 block, 256 threads (8 waves).
// A frag (16x4 f32): lane l -> M = l%16, K = kb + 2*(l/16) + {0,1} in v0,v1
// B frag (4x16 f32): lane l -> N = l%16, K = kb + 2*(l/16) + {0,1} in v0,v1
// D frag (16x16 f32): vgpr v, lane l -> (M = v + 8*(l/16), N = l%16)
// ---------------------------------------------------------------------------
__global__ void se_mlp_kernel(const float* __restrict__ partial,
                              const int* __restrict__ counts,
                              const float* __restrict__ W1, const float* __restrict__ b1,
                              const float* __restrict__ W2, const float* __restrict__ b2,
                              float* __restrict__ g_y) {            // [NSEG][C]
    __shared__ float s_pooled[NSEG * C];   // 32 KB
    __shared__ float s_h[NSEG * CR];       // 2 KB
    __shared__ float s_cnt[NSEG];

    const int tid  = threadIdx.x;
    const int wid  = tid >> 5;
    const int lane = tid & 31;
    const int half = lane >> 4;
    const int l16  = lane & 15;

    if (tid < NSEG) {
        int cnt = 0;
        for (int s = 0; s < SPLITS; ++s) cnt += counts[tid * SPLITS + s];
        s_cnt[tid] = fmaxf((float)cnt, 1.0f);
    }
    __syncthreads();

    // pooled mean: fixed-order reduction over splits (deterministic)
    for (int seg = 0; seg < NSEG; ++seg) {
        float sum = 0.0f;
        for (int s = 0; s < SPLITS; ++s)
            sum += partial[((size_t)seg * SPLITS + s) * C + tid];
        s_pooled[seg * C + tid] = sum / s_cnt[seg];
    }
    __syncthreads();

    // ---- GEMM1: [32,256] x [256,16] -> h[32,16]; waves 0,1 own M-tiles ----
    if (wid < 2) {                                 // wave-uniform: EXEC stays all-1s
        const int mt = wid;
        const int m  = mt * 16 + l16;
        v8f acc = {0.f, 0.f, 0.f, 0.f, 0.f, 0.f, 0.f, 0.f};
        for (int k = 0; k < C; k += 4) {
            const int ka = k + 2 * half;
            v2f a; a.x = s_pooled[m * C + ka];      a.y = s_pooled[m * C + ka + 1];
            v2f b; b.x = W1[ka * CR + l16];         b.y = W1[(ka + 1) * CR + l16];
            acc = __builtin_amdgcn_wmma_f32_16x16x4_f32(false, a, false, b,
                                                        (short)0, acc, false, false);
        }
        const float bias = b1[l16];
#pragma unroll
        for (int v = 0; v < 8; ++v) {
            const int row = mt * 16 + v + 8 * half;
            s_h[row * CR + l16] = fmaxf(acc[v] + bias, 0.0f);   // ReLU
        }
    }
    __syncthreads();

    // ---- GEMM2: [32,16] x [16,256] -> y[32,256]; 32 tiles over 8 waves ----
    for (int t = wid * 4; t < wid * 4 + 4; ++t) {
        const int mt = t >> 4;        // 0..1
        const int nt = t & 15;        // 0..15
        v8f acc = {0.f, 0.f, 0.f, 0.f, 0.f, 0.f, 0.f, 0.f};
        for (int kb = 0; kb < CR; kb += 4) {
            const int ka = kb + 2 * half;
            v2f a; a.x = s_h[(mt * 16 + l16) * CR + ka];
                   a.y = s_h[(mt * 16 + l16) * CR + ka + 1];
            v2f b; b.x = W2[ka * C + nt * 16 + l16];
                   b.y = W2[(ka + 1) * C + nt * 16 + l16];
            acc = __builtin_amdgcn_wmma_f32_16x16x4_f32(false, a, false, b,
                                                        (short)0, acc, false, false);
        }
        const int col  = nt * 16 + l16;
        const float bc = b2[col];
#pragma unroll
        for (int v = 0; v < 8; ++v) {
            const int row = mt * 16 + v + 8 * half;
            const float x = acc[v] + bc;
            g_y[row * C + col] = 1.0f / (1.0f + __expf(-x));    // sigmoid
        }
    }
}

// ---------------------------------------------------------------------------
// Kernel 3: out[r,:] = feats[r,:] * y[batch_idx[r],:]. Gate table in LDS,
// float4 streaming (64 lanes/row, 4 rows per block-iteration).
// ---------------------------------------------------------------------------
__global__ void scale_kernel(const float4* __restrict__ feats4,
                             const int* __restrict__ bidx,
                             const float4* __restrict__ g_y4,
                             float4* __restrict__ out4, int N) {
    __shared__ float4 s_y[NSEG * (C / 4)];   // 32 KB gate table
    for (int i = threadIdx.x; i < NSEG * (C / 4); i += blockDim.x)
        s_y[i] = g_y4[i];
    __syncthreads();

    const int quad = threadIdx.x & 63;   // column group (4 floats each)
    const int rl   = threadIdx.x >> 6;   // 0..3: row within block-iteration
    const int rstep = gridDim.x * 4;

    for (int r = blockIdx.x * 4 + rl; r < N; r += rstep) {
        __builtin_prefetch(&feats4[(size_t)(r + rstep) * 64 + quad], 0, 0);
        const int seg  = bidx[r];
        const float4 f = feats4[(size_t)r * 64 + quad];
        const float4 g = s_y[seg * 64 + quad];
        float4 o;
        o.x = f.x * g.x; o.y = f.y * g.y; o.z = f.z * g.z; o.w = f.w * g.w;
        out4[(size_t)r * 64 + quad] = o;
    }
}

// ---------------------------------------------------------------------------
extern "C" void kernel_launch(void* const* d_in, const int* in_sizes, int n_in,
                              void* d_out, int out_size, void* d_ws, size_t ws_size,
                              hipStream_t stream) {
    const float* feats = (const float*)d_in[0];
    const int*   bidx  = (const int*)d_in[1];
    const float* W1    = (const float*)d_in[2];
    const float* b1    = (const float*)d_in[3];
    const float* W2    = (const float*)d_in[4];
    const float* b2    = (const float*)d_in[5];
    float*       out   = (float*)d_out;
    const int N = in_sizes[1];              // number of points

    // workspace layout
    float* partial = (float*)d_ws;                                     // 2 MB
    int*   counts  = (int*)((char*)d_ws + (size_t)NSEG * SPLITS * C * 4);
    float* g_y     = (float*)((char*)counts + (size_t)NSEG * SPLITS * 4);

    pool_partial_kernel<<<NSEG * SPLITS, 256, 0, stream>>>(feats, bidx, N, partial, counts);
    se_mlp_kernel<<<1, 256, 0, stream>>>(partial, counts, W1, b1, W2, b2, g_y);
    scale_kernel<<<4096, 256, 0, stream>>>((const float4*)feats, bidx,
                                           (const float4*)g_y, (float4*)out, N);
}